// PanopticSegment_12103217840194
// MI455X (gfx1250) — compile-verified
//
#include <hip/hip_runtime.h>
#include <math.h>

#define BB 4
#define CC 4
#define HH 160
#define WW 160
#define NPIX (HH*WW)          // 25600
#define PRE_K 2048
#define TOPK 500
#define NBOX 512              // TOPK padded to WMMA tile multiple
#define IOU_T 0.5f
#define PROB_T 0.75f
#define COV2 100.0f           // COV^2
#define NEG_HALF_LOG2E (-0.7213475204444817f)   // -0.5 * log2(e)
#define NEG_BIG (-3.0e38f)                      // "log2(0)" sentinel -> exp2 == 0

// output layout (floats): heat[B*NPIX] | mask[B*C*NPIX] | fboxes[B*TOPK*4] | fvalid[B*TOPK]
#define OUT_MASK   (BB*NPIX)                   // 102400
#define OUT_BOXES  (OUT_MASK + BB*CC*NPIX)     // 512000
#define OUT_VALID  (OUT_BOXES + BB*TOPK*4)     // 520000

typedef float v2f __attribute__((ext_vector_type(2)));
typedef float v8f __attribute__((ext_vector_type(8)));
typedef unsigned long long u64;
typedef unsigned int u32;

// ---------------- kernel 1: softmax / class / score / box decode ----------------
__global__ void k_scores(const float* __restrict__ mask, const float* __restrict__ bias,
                         float* __restrict__ scoreG, float* __restrict__ boxG) {
  int t = blockIdx.x * blockDim.x + threadIdx.x;
  if (t >= BB * NPIX) return;
  int b = t / NPIX, p = t % NPIX;
  int y = p / WW, x = p % WW;
  const float* m = mask + (size_t)b * CC * NPIX + p;
  float v0 = m[0], v1 = m[NPIX], v2 = m[2 * NPIX], v3 = m[3 * NPIX];
  float mx = fmaxf(fmaxf(v0, v1), fmaxf(v2, v3));
  float s = __expf(v0 - mx) + __expf(v1 - mx) + __expf(v2 - mx) + __expf(v3 - mx);
  float prob = 1.0f / s;                 // max softmax = exp(0)/sum
  int cls = 0; float best = v0;
  if (v1 > best) { best = v1; cls = 1; }
  if (v2 > best) { best = v2; cls = 2; }
  if (v3 > best) { best = v3; cls = 3; }
  bool frontal = (cls != 0) && (prob > PROB_T);
  scoreG[t] = frontal ? prob : -1.0f;
  const float* bi = bias + (size_t)b * 4 * NPIX + p;
  float* bo = boxG + (size_t)t * 4;
  bo[0] = (float)y + bi[0];
  bo[1] = (float)x + bi[NPIX];
  bo[2] = (float)y + bi[2 * NPIX];
  bo[3] = (float)x + bi[3 * NPIX];
}

// ---------------- kernel 2: mask passthrough ----------------
__global__ void k_copy(const float* __restrict__ in, float* __restrict__ out, int n) {
  int t = blockIdx.x * blockDim.x + threadIdx.x;
  if (t < n) out[t] = in[t];
}

// ---------------- kernel 3: per-image bitonic sort (top-2048 sorted desc) ----------------
// keys: (order-preserving score bits << 32) | (~idx) so ties break ascending-index like top_k.
__global__ void k_sort(const float* __restrict__ scoreG, int* __restrict__ sIdx,
                       float* __restrict__ sScore) {
  extern __shared__ u64 key[];           // 32768 * 8B = 256 KB dynamic LDS (<=320KB/WGP)
  const int N = 32768;
  int b = blockIdx.x, tid = threadIdx.x, nt = blockDim.x;
  for (int i = tid; i < N; i += nt) {
    u64 k = 0ull;
    if (i < NPIX) {
      u32 bits = __float_as_uint(scoreG[b * NPIX + i]);
      u32 o = (bits & 0x80000000u) ? ~bits : (bits | 0x80000000u);
      k = ((u64)o << 32) | (u64)(0xFFFFFFFFu - (u32)i);
    }
    key[i] = k;
  }
  __syncthreads();
  for (int kk = 2; kk <= N; kk <<= 1) {
    for (int j = kk >> 1; j > 0; j >>= 1) {
      for (int i = tid; i < N; i += nt) {
        int ij = i ^ j;
        if (ij > i) {
          u64 a = key[i], c = key[ij];
          bool desc = ((i & kk) == 0);
          if ((a < c) == desc) { key[i] = c; key[ij] = a; }
        }
      }
      __syncthreads();
    }
  }
  for (int i = tid; i < PRE_K; i += nt) {
    int idx = (int)(0xFFFFFFFFu - (u32)(key[i] & 0xFFFFFFFFull));
    sIdx[b * PRE_K + i] = idx;
    sScore[b * PRE_K + i] = scoreG[b * NPIX + idx];
  }
}

// ---------------- kernel 4: greedy NMS, LDS-resident ----------------
__global__ void k_nms(const float* __restrict__ boxG, const int* __restrict__ sIdx,
                      const float* __restrict__ sScore, int* __restrict__ keepG) {
  __shared__ float bx[PRE_K][4];         // 32KB
  __shared__ float area[PRE_K];          // 8KB
  __shared__ int keep[PRE_K];            // 8KB
  int b = blockIdx.x, tid = threadIdx.x, nt = blockDim.x;
  for (int j = tid; j < PRE_K; j += nt) {
    int idx = sIdx[b * PRE_K + j];
    const float* p = boxG + ((size_t)b * NPIX + idx) * 4;
    float y1 = p[0], x1 = p[1], y2 = p[2], x2 = p[3];
    bx[j][0] = y1; bx[j][1] = x1; bx[j][2] = y2; bx[j][3] = x2;
    area[j] = (y2 - y1) * (x2 - x1);     // unclamped, per reference
    keep[j] = (sScore[b * PRE_K + j] > 0.0f) ? 1 : 0;
  }
  __syncthreads();
  for (int i = 0; i < PRE_K; ++i) {
    if (keep[i]) {
      float iy1 = bx[i][0], ix1 = bx[i][1], iy2 = bx[i][2], ix2 = bx[i][3], ia = area[i];
      for (int j = i + 1 + tid; j < PRE_K; j += nt) {
        if (keep[j]) {
          float h = fmaxf(fminf(iy2, bx[j][2]) - fmaxf(iy1, bx[j][0]), 0.0f);
          float w = fmaxf(fminf(ix2, bx[j][3]) - fmaxf(ix1, bx[j][1]), 0.0f);
          float inter = h * w;
          float iou = inter / (ia + area[j] - inter);  // NaN compares false, matches jnp
          if (iou > IOU_T) keep[j] = 0;
        }
      }
    }
    __syncthreads();
  }
  for (int j = tid; j < PRE_K; j += nt) keepG[b * PRE_K + j] = keep[j];
}

// ---------------- kernel 5: final selection + folded-exponent coefficients ----------------
// coef row n: D[n,p] = -0.5*log2e*z^2[n,p] + log2(prob_n), so heat = exp2(max D).
__global__ void k_final(const float* __restrict__ boxG, const int* __restrict__ sIdx,
                        const float* __restrict__ sScore, const int* __restrict__ keepG,
                        float* __restrict__ coefG, float* __restrict__ fprobG,
                        float* __restrict__ out) {
  __shared__ int fi[NBOX];
  int b = blockIdx.x, n = threadIdx.x;
  if (n == 0) {                          // top_k over kept_s: kept (sorted) first, then -1 ties by index
    int cnt = 0;
    for (int i = 0; i < PRE_K && cnt < NBOX; ++i) if (keepG[b * PRE_K + i]) fi[cnt++] = i;
    for (int i = 0; i < PRE_K && cnt < NBOX; ++i) if (!keepG[b * PRE_K + i]) fi[cnt++] = i;
  }
  __syncthreads();
  if (n >= NBOX) return;
  const float s = NEG_HALF_LOG2E;
  float c0, c1, c2, c3, c4, prob;
  if (n < TOPK) {
    int ci = fi[n];
    int idx = sIdx[b * PRE_K + ci];
    const float* p = boxG + ((size_t)b * NPIX + idx) * 4;
    float b0 = p[0], b1 = p[1], b2 = p[2], b3 = p[3];
    int valid = keepG[b * PRE_K + ci];
    prob = valid ? sScore[b * PRE_K + ci] : 0.0f;
    float my = (b0 + b2) * 0.5f, mxx = (b1 + b3) * 0.5f;
    float sy = valid ? (b2 - b0) : 1.0f;
    float sx = valid ? (b3 - b1) : 1.0f;
    float ay = COV2 / (sy * sy), ax = COV2 / (sx * sx);
    float lprob = valid ? __log2f(prob) : NEG_BIG;
    c0 = s * ay; c1 = -2.0f * (s * ay) * my;
    c2 = s * ax; c3 = -2.0f * (s * ax) * mxx;
    c4 = s * (ay * my * my + ax * mxx * mxx) + lprob;
    float* ob = out + OUT_BOXES + ((size_t)b * TOPK + n) * 4;
    ob[0] = b0; ob[1] = b1; ob[2] = b2; ob[3] = b3;
    out[OUT_VALID + b * TOPK + n] = valid ? 1.0f : 0.0f;
  } else {                               // WMMA padding rows: exponent pinned at -huge -> exp2 == 0
    prob = 0.0f; c0 = s; c1 = 0.0f; c2 = s; c3 = 0.0f; c4 = NEG_BIG;
  }
  float* cf = coefG + ((size_t)b * NBOX + n) * 8;
  cf[0] = c0; cf[1] = c1; cf[2] = c2; cf[3] = c3; cf[4] = c4;
  cf[5] = 0.0f; cf[6] = 0.0f; cf[7] = 0.0f;
  fprobG[b * NBOX + n] = prob;
}

// ---------------- kernel 6: WMMA gaussian heatmap, exponent-domain max ----------------
// D[n,p] = A[n,0:8] . F[0:8,p] via two chained V_WMMA_F32_16X16X4_F32 (K=8); one exp2/pixel.
// A layout per ISA: lanes 0-15 hold {K0,K1}, lanes 16-31 hold {K2,K3} of row M=lane&15.
__global__ void __launch_bounds__(256) k_heat(const float* __restrict__ coefG,
                                              const float* __restrict__ fprobG,
                                              float* __restrict__ out) {
  __shared__ float cf[NBOX * 8];         // 16KB box coefficients
  __shared__ u32 tmask;                  // bit m = 16-box tile m has any prob>0
  int b = blockIdx.y, tid = threadIdx.x;
  __builtin_prefetch(coefG + (size_t)b * NBOX * 8, 0, 1);   // global_prefetch_b8
  for (int i = tid; i < NBOX * 8; i += 256) cf[i] = coefG[(size_t)b * NBOX * 8 + i];
  if (tid < 32) {                        // wave 0: one lane per box tile, ballot packs 32 flags
    int f = 0;
    for (int j = 0; j < 16; ++j) f |= (fprobG[b * NBOX + tid * 16 + j] > 0.0f) ? 1 : 0;
    u32 bal = (u32)__ballot(f);
    if (tid == 0) tmask = bal;
  }
  __syncthreads();
  int wave = tid >> 5, lane = tid & 31;
  int tile = blockIdx.x * 8 + wave;      // 1600 tiles of 16 pixels per image
  int n = lane & 15, hi = lane >> 4;
  int p = tile * 16 + n;                 // WW divisible by 16: tile stays in one row
  float py = (float)(p / WW), px = (float)(p % WW);
  // B operand: K0=py^2 K1=py K2=px^2 K3=px | K4=1 K5..7=0
  float f0 = (hi == 0) ? py * py : px * px;
  float f1 = (hi == 0) ? py : px;
  v2f bop0 = {f0, f1};
  v2f bop1 = {(hi == 0) ? 1.0f : 0.0f, 0.0f};
  int klo = hi * 2;
  u32 am = (u32)__builtin_amdgcn_readfirstlane((int)tmask);  // SGPR mask, loaded once
  float hmax = NEG_BIG;                  // exponent-domain identity
#pragma unroll
  for (int m = 0; m < NBOX / 16; ++m) {
    if (am & (1u << m)) {                // pure SALU bit test: EXEC stays all-1s
      const float* c = &cf[(m * 16 + n) * 8];
      v2f a0 = {c[klo], c[klo + 1]};
      v2f a1 = {c[4 + klo], c[4 + klo + 1]};
      v8f acc = {0.f, 0.f, 0.f, 0.f, 0.f, 0.f, 0.f, 0.f};
      acc = __builtin_amdgcn_wmma_f32_16x16x4_f32(false, a0, false, bop0, (short)0, acc, false, false);
      acc = __builtin_amdgcn_wmma_f32_16x16x4_f32(false, a1, false, bop1, (short)0, acc, false, false);
#pragma unroll
      for (int v = 0; v < 8; ++v) hmax = fmaxf(hmax, acc[v]);   // -> v_max3_num_f32 pairs
    }
  }
  float other = __shfl_xor(hmax, 16, 32);  // fold M=0-7 / M=8-15 lane halves
  hmax = fmaxf(hmax, other);
  if (hi == 0) out[b * NPIX + p] = exp2f(hmax);   // single v_exp_f32 per pixel
}

extern "C" void kernel_launch(void* const* d_in, const int* in_sizes, int n_in,
                              void* d_out, int out_size, void* d_ws, size_t ws_size,
                              hipStream_t stream) {
  const float* mask = (const float*)d_in[0];
  const float* bias = (const float*)d_in[1];
  float* out = (float*)d_out;
  float* ws = (float*)d_ws;
  // workspace layout (floats)
  float* scoreG = ws;                         // 102400
  float* boxG   = ws + 102400;                // 409600
  int*   sIdx   = (int*)(ws + 512000);        // 8192
  int*   keepG  = (int*)(ws + 520192);        // 8192
  float* sScore = ws + 528384;                // 8192
  float* coefG  = ws + 536576;                // 16384
  float* fprobG = ws + 552960;                // 2048  (total ~2.2 MB)

  k_scores<<<dim3((BB * NPIX + 255) / 256), 256, 0, stream>>>(mask, bias, scoreG, boxG);
  k_copy<<<dim3((BB * CC * NPIX + 255) / 256), 256, 0, stream>>>(mask, out + OUT_MASK, BB * CC * NPIX);
  k_sort<<<BB, 1024, 32768 * sizeof(unsigned long long), stream>>>(scoreG, sIdx, sScore);
  k_nms<<<BB, 1024, 0, stream>>>(boxG, sIdx, sScore, keepG);
  k_final<<<BB, NBOX, 0, stream>>>(boxG, sIdx, sScore, keepG, coefG, fprobG, out);
  k_heat<<<dim3(200, BB), 256, 0, stream>>>(coefG, fprobG, out);
}